// GlobalModel_80032420593875
// MI455X (gfx1250) — compile-verified
//
#include <hip/hip_runtime.h>

typedef float v2f __attribute__((ext_vector_type(2)));
typedef float v8f __attribute__((ext_vector_type(8)));

// ---------------------------------------------------------------------------
// Problem shape (fixed by the reference)
// ---------------------------------------------------------------------------
#define D_NODE 128
#define D_EDGE 64
#define D_GLOB 128
#define D_IN   320   // D_GLOB + D_NODE + D_EDGE
#define D_HID  256
#define D_OUT  128
#define NSEG   1024  // number of graphs (segments)
#define SLICE_COLS 64

// ---------------------------------------------------------------------------
// Kernel 1: zero final accumulators in workspace
// ---------------------------------------------------------------------------
__global__ __launch_bounds__(256) void zero_kernel(float* __restrict__ p, int n) {
    int i = blockIdx.x * 256 + threadIdx.x;
    if (i < n) p[i] = 0.0f;
}

// ---------------------------------------------------------------------------
// Kernel 2/3: LDS-privatized scatter-add.
// Each block owns (column-slice of 64 cols, row-chunk). The full 1024-segment
// accumulator for that slice lives in dynamic LDS (1024*64*4B = 256KB + 4KB
// counts, within the 320KB/WGP CDNA5 limit). Streaming loads are coalesced
// (256B contiguous per row); per-element adds hit ds_add_f32 (LDS atomics),
// and only one global fp32 atomic per LDS entry is issued at flush time
// (34M total vs 256M element-wise).
// ---------------------------------------------------------------------------
__global__ __launch_bounds__(256) void scatter_lds_kernel(
    const float* __restrict__ feats, const int* __restrict__ seg,
    float* __restrict__ sums, float* __restrict__ cnt,
    int nrows, int row_stride, int chunks, int chunk_rows) {
    extern __shared__ float smem[];                 // [NSEG*64] acc + [NSEG] counts
    float* acc  = smem;
    float* accc = smem + NSEG * SLICE_COLS;

    const int bx    = blockIdx.x;
    const int slice = bx / chunks;                  // which 64-col slice
    const int chunk = bx - slice * chunks;          // which row chunk
    const int tid   = threadIdx.x;
    const bool count_here = (slice == 0);

    // zero LDS accumulator
    for (int i = tid; i < NSEG * SLICE_COLS + NSEG; i += 256) smem[i] = 0.0f;
    __syncthreads();

    // 256 threads = 16 rows in flight x 16 float4 columns
    const int r_off = tid >> 4;                     // 0..15
    const int c4    = tid & 15;                     // float4 slot in slice
    const int col0  = slice * SLICE_COLS + c4 * 4;
    long row_end = (long)(chunk + 1) * chunk_rows;
    if (row_end > nrows) row_end = nrows;

    for (long row = (long)chunk * chunk_rows + r_off; row < row_end; row += 16) {
        const int s = seg[row];
        const float4 v = *(const float4*)(feats + row * row_stride + col0);
        float* p = acc + s * SLICE_COLS + c4 * 4;
        unsafeAtomicAdd(p + 0, v.x);                // ds_add_f32
        unsafeAtomicAdd(p + 1, v.y);
        unsafeAtomicAdd(p + 2, v.z);
        unsafeAtomicAdd(p + 3, v.w);
        if (count_here && c4 == 0) unsafeAtomicAdd(accc + s, 1.0f);
    }
    __syncthreads();

    // flush LDS accumulator to global with one atomic per entry
    for (int i = tid; i < NSEG * SLICE_COLS; i += 256) {
        const float v = acc[i];
        if (v != 0.0f)
            unsafeAtomicAdd(sums + (long)(i >> 6) * row_stride +
                                   slice * SLICE_COLS + (i & 63), v);
    }
    if (count_here) {
        for (int i = tid; i < NSEG; i += 256) {
            const float v = accc[i];
            if (v != 0.0f) unsafeAtomicAdd(cnt + i, v);
        }
    }
}

// ---------------------------------------------------------------------------
// Kernel 4: build concat rows  [u | node_sum/max(cnt,1) | edge_sum/max(cnt,1)]
// ---------------------------------------------------------------------------
__global__ __launch_bounds__(256) void concat_kernel(
    const float* __restrict__ u,
    const float* __restrict__ node_sum, const float* __restrict__ node_cnt,
    const float* __restrict__ edge_sum, const float* __restrict__ edge_cnt,
    float* __restrict__ concat, int B) {
    int t = blockIdx.x * 256 + threadIdx.x;
    if (t >= B * D_IN) return;
    int g = t / D_IN;
    int c = t - g * D_IN;
    float v;
    if (c < D_GLOB) {
        v = u[g * D_GLOB + c];
    } else if (c < D_GLOB + D_NODE) {
        v = node_sum[g * D_NODE + (c - D_GLOB)] / fmaxf(node_cnt[g], 1.0f);
    } else {
        v = edge_sum[g * D_EDGE + (c - D_GLOB - D_NODE)] / fmaxf(edge_cnt[g], 1.0f);
    }
    concat[t] = v;
}

// ---------------------------------------------------------------------------
// Kernel 5: fused 2-layer MLP with V_WMMA_F32_16X16X4_F32 (exact fp32).
// Block = 256 threads (8 waves), one 16-row tile of the 1024-row batch.
// Stage 1: wave w -> hidden cols [32w, 32w+32), ReLU into LDS (pad=4 ->
//          conflict-free stage-2 A reads: bank = (4*row + col) % 64).
// Stage 2: wave w -> output cols [16w, 16w+16).
// ---------------------------------------------------------------------------
#define HPAD (D_HID + 4)

__global__ __launch_bounds__(256) void mlp_kernel(
    const float* __restrict__ concat,
    const float* __restrict__ W1, const float* __restrict__ b1,
    const float* __restrict__ W2, const float* __restrict__ b2,
    float* __restrict__ out) {
    __shared__ float Hs[16 * HPAD];

    const int tid  = threadIdx.x;
    const int lane = tid & 31;
    const int wave = tid >> 5;
    const int l    = lane & 15;   // column / M-row selector within half
    const int h    = lane >> 4;   // half select (K/M offset per ISA layout)
    const int row0 = blockIdx.x * 16;

    // ---------------- Stage 1: H = relu(concat @ W1 + b1) ----------------
    #pragma unroll
    for (int t = 0; t < 2; ++t) {
        const int col0 = wave * 32 + t * 16;
        v8f c = {};
        for (int k0 = 0; k0 < D_IN; k0 += 4) {
            // A 16x4 fragment: VGPR j holds K = k0 + j + 2*h, M = row0 + l
            const float* ap = concat + (long)(row0 + l) * D_IN + k0 + 2 * h;
            v2f a; a.x = ap[0]; a.y = ap[1];
            // B 4x16 fragment: VGPR j holds K = k0 + j + 2*h, N = col0 + l
            const float* bp = W1 + (long)(k0 + 2 * h) * D_HID + col0 + l;
            v2f b; b.x = bp[0]; b.y = bp[D_HID];
            c = __builtin_amdgcn_wmma_f32_16x16x4_f32(false, a, false, b,
                                                      (short)0, c, false, false);
        }
        const float bias = b1[col0 + l];
        #pragma unroll
        for (int i = 0; i < 8; ++i) {
            float v = c[i] + bias;               // C: VGPR i -> M = i + 8*h
            v = v > 0.0f ? v : 0.0f;
            Hs[(i + 8 * h) * HPAD + col0 + l] = v;
        }
    }
    __syncthreads();

    // ---------------- Stage 2: out = H @ W2 + b2 ----------------
    {
        const int col0 = wave * 16;
        v8f c = {};
        for (int k0 = 0; k0 < D_HID; k0 += 4) {
            const float* ap = Hs + l * HPAD + k0 + 2 * h;
            v2f a; a.x = ap[0]; a.y = ap[1];
            const float* bp = W2 + (long)(k0 + 2 * h) * D_OUT + col0 + l;
            v2f b; b.x = bp[0]; b.y = bp[D_OUT];
            c = __builtin_amdgcn_wmma_f32_16x16x4_f32(false, a, false, b,
                                                      (short)0, c, false, false);
        }
        const float bias = b2[col0 + l];
        #pragma unroll
        for (int i = 0; i < 8; ++i) {
            out[(long)(row0 + i + 8 * h) * D_OUT + col0 + l] = c[i] + bias;
        }
    }
}

// ---------------------------------------------------------------------------
// Host launcher
// ---------------------------------------------------------------------------
extern "C" void kernel_launch(void* const* d_in, const int* in_sizes, int n_in,
                              void* d_out, int out_size, void* d_ws, size_t ws_size,
                              hipStream_t stream) {
    const float* x         = (const float*)d_in[0];
    const float* edge_attr = (const float*)d_in[1];
    const float* u         = (const float*)d_in[2];
    const int*   v_idx     = (const int*)d_in[3];
    const int*   e_idx     = (const int*)d_in[4];
    const float* W1        = (const float*)d_in[5];
    const float* b1        = (const float*)d_in[6];
    const float* W2        = (const float*)d_in[7];
    const float* b2        = (const float*)d_in[8];
    float*       out       = (float*)d_out;

    const int N = in_sizes[0] / D_NODE;   // 1,000,000
    const int E = in_sizes[1] / D_EDGE;   // 2,000,000
    const int B = in_sizes[2] / D_GLOB;   // 1024

    // Workspace layout (floats), ~2.1 MB total
    float* ws       = (float*)d_ws;
    float* node_sum = ws;                                   // B*D_NODE
    float* edge_sum = node_sum + (long)B * D_NODE;          // B*D_EDGE
    float* node_cnt = edge_sum + (long)B * D_EDGE;          // B
    float* edge_cnt = node_cnt + B;                         // B
    float* concat   = edge_cnt + B;                         // B*D_IN
    const int n_accum = B * D_NODE + B * D_EDGE + 2 * B;

    const size_t lds_bytes = (size_t)(NSEG * SLICE_COLS + NSEG) * sizeof(float);

    // 1) zero accumulators (must happen every call; harness doesn't re-zero)
    zero_kernel<<<(n_accum + 255) / 256, 256, 0, stream>>>(node_sum, n_accum);

    // 2) node scatter: 2 column slices x 128 row chunks = 256 big-LDS blocks
    {
        const int chunks = 128;
        const int chunk_rows = (N + chunks - 1) / chunks;
        scatter_lds_kernel<<<(D_NODE / SLICE_COLS) * chunks, 256, lds_bytes, stream>>>(
            x, v_idx, node_sum, node_cnt, N, D_NODE, chunks, chunk_rows);
    }
    // 3) edge scatter: 1 column slice x 256 row chunks = 256 big-LDS blocks
    {
        const int chunks = 256;
        const int chunk_rows = (E + chunks - 1) / chunks;
        scatter_lds_kernel<<<(D_EDGE / SLICE_COLS) * chunks, 256, lds_bytes, stream>>>(
            edge_attr, e_idx, edge_sum, edge_cnt, E, D_EDGE, chunks, chunk_rows);
    }
    // 4) concat rows
    {
        int total = B * D_IN;
        concat_kernel<<<(total + 255) / 256, 256, 0, stream>>>(
            u, node_sum, node_cnt, edge_sum, edge_cnt, concat, B);
    }
    // 5) fused MLP with fp32 WMMA
    mlp_kernel<<<B / 16, 256, 0, stream>>>(concat, W1, b1, W2, b2, out);
}